// HisCoM_4200478015920
// MI455X (gfx1250) — compile-verified
//
#include <hip/hip_runtime.h>
#include <math.h>

// Problem constants (match reference)
#define BB 4096
#define GG 512
#define DD 64
#define PP (GG * DD)       // 32768 features
#define EPS 1e-5f
#define NEG_SLOPE 0.2f

typedef __attribute__((ext_vector_type(2))) float v2f;
typedef __attribute__((ext_vector_type(8))) float v8f;
typedef __attribute__((ext_vector_type(4))) int   v4i;

// LDS x-tile row stride: 512 cols + 4 pad floats -> bank-conflict-free b64 reads
#define XROW 516

#if defined(__has_builtin)
#if __has_builtin(__builtin_amdgcn_global_load_async_to_lds_b128)
#define HAVE_ASYNC_LDS 1
#endif
#endif

#ifdef HAVE_ASYNC_LDS
// builtin signature: (global v4i* src, local v4i* dst, imm offset, imm cpol)
typedef __attribute__((address_space(1))) v4i* gas_ptr_t;   // global b128
typedef __attribute__((address_space(3))) v4i* las_ptr_t;   // LDS b128
#endif

static __device__ __forceinline__ void wait_async_zero() {
#ifdef HAVE_ASYNC_LDS
#if __has_builtin(__builtin_amdgcn_s_wait_asynccnt)
    __builtin_amdgcn_s_wait_asynccnt(0);
#else
    asm volatile("s_wait_asynccnt 0" ::: "memory");
#endif
#endif
}

// ---------------------------------------------------------------------------
// Kernel 1: reps tile via WMMA f32 16x16x4, LeakyReLU, stat atomics, store
// grid = (G/8, B/16), block = 256 (8 waves). Wave w handles group g0+w.
// ---------------------------------------------------------------------------
__global__ void hiscom_reps_kernel(const float* __restrict__ x,
                                   const int* __restrict__ idx,
                                   const float* __restrict__ W,
                                   float* __restrict__ repsT,   // [G][B]
                                   float* __restrict__ gsum,    // [G]
                                   float* __restrict__ gsumsq)  // [G]
{
    __shared__ __align__(16) float xs[16 * XROW];   // x tile 16 batches x 512 feats (+pad)
    __shared__ __align__(16) float wl[8 * DD];      // weights for 8 groups

    const int t  = threadIdx.x;
    const int g0 = blockIdx.x * 8;          // first group of this block
    const int b0 = blockIdx.y * 16;         // first batch row of this tile
    const int fbase = g0 * DD;              // first feature column (block owns 512 cols)

    // ---- stage W[g0..g0+7][0..63] -> LDS (512 floats) ----
    wl[t]       = W[fbase + t];
    wl[t + 256] = W[fbase + t + 256];

    // ---- stage x tile (16 rows x 512 cols) -> LDS, honoring idx gather ----
    // 2048 float4-chunks, 8 per thread; coalesced along features.
    // Fast path (consecutive, aligned indices) goes straight to LDS via the
    // gfx1250 async-to-LDS path (ASYNCcnt); fallback gathers through VGPRs.
    const int4* idx4 = reinterpret_cast<const int4*>(idx + fbase);
#pragma unroll
    for (int k = 0; k < 8; ++k) {
        int cid = t + k * 256;              // chunk id 0..2047
        int row = cid >> 7;                 // 128 chunks per row
        int c4  = cid & 127;
        const float* xr = x + (size_t)(b0 + row) * PP;
        float* dstl = &xs[row * XROW + c4 * 4];
        int4 iv = idx4[c4];
        if (iv.y == iv.x + 1 && iv.z == iv.x + 2 && iv.w == iv.x + 3 &&
            ((iv.x & 3) == 0)) {
#ifdef HAVE_ASYNC_LDS
            __builtin_amdgcn_global_load_async_to_lds_b128(
                (gas_ptr_t)(xr + iv.x), (las_ptr_t)dstl, 0, 0);
#else
            *reinterpret_cast<float4*>(dstl) =
                *reinterpret_cast<const float4*>(xr + iv.x);
#endif
        } else {
            dstl[0] = xr[iv.x];
            dstl[1] = xr[iv.y];
            dstl[2] = xr[iv.z];
            dstl[3] = xr[iv.w];
        }
    }
    wait_async_zero();                      // async LDS writes visible pre-barrier
    __syncthreads();

    // ---- per-wave WMMA chain: group g = g0 + wave, 16 batches, K = 64 ----
    const int wave = t >> 5;
    const int lane = t & 31;
    const int r = lane & 15;                // A-matrix row (batch within tile)
    const int h = lane >> 4;                // lane-half: selects K pair {0,1} vs {2,3}
    const int g = g0 + wave;

    const float* arow = &xs[r * XROW + 2 * h];
    const float* brow = &wl[wave * DD + 2 * h];

    v8f c = {};
#pragma unroll
    for (int s = 0; s < 16; ++s) {          // K steps of 4: d0 = 4*s
        v2f a = *reinterpret_cast<const v2f*>(arow + 4 * s);
        v2f b = *reinterpret_cast<const v2f*>(brow + 4 * s);
        c = __builtin_amdgcn_wmma_f32_16x16x4_f32(false, a, false, b,
                                                  (short)0, c, false, false);
    }

    // ---- LeakyReLU(0.2) on accumulators; partial batch sums ----
    float s1 = 0.f, s2 = 0.f;
#pragma unroll
    for (int i = 0; i < 8; ++i) {
        float v = c[i];
        v = (v >= 0.f) ? v : NEG_SLOPE * v;
        c[i] = v;
        s1 += v;
        s2 += v * v;
    }
    // combine the two lane-halves (M = i + 8*h) -> full 16-batch partial
    s1 += __shfl_xor(s1, 16, 32);
    s2 += __shfl_xor(s2, 16, 32);
    if (lane == 0) {
        atomicAdd(&gsum[g], s1);
        atomicAdd(&gsumsq[g], s2);
    }

    // ---- store reps transposed [G][B]; columns of D are replicated, take n==0 ----
    if ((lane & 15) == 0) {
        float* dst = repsT + (size_t)g * BB + b0 + 8 * h;
#pragma unroll
        for (int i = 0; i < 8; ++i) dst[i] = c[i];
    }
}

// ---------------------------------------------------------------------------
// Kernel 2: finalize batch-norm stats, fold gamma/beta/fc_w into a[] and K
// one block of 512 threads (thread t = group t)
// ---------------------------------------------------------------------------
__global__ void hiscom_stats_kernel(const float* __restrict__ gsum,
                                    const float* __restrict__ gsumsq,
                                    const float* __restrict__ gamma,
                                    const float* __restrict__ beta,
                                    const float* __restrict__ fc_w,
                                    const float* __restrict__ fc_b,
                                    float* __restrict__ avec,   // [G]
                                    float* __restrict__ kons)   // [1]
{
    __shared__ float red[512];
    const int t = threadIdx.x;
    const float invB = 1.0f / (float)BB;
    float mean = gsum[t] * invB;
    float var  = gsumsq[t] * invB - mean * mean;
    float inv  = rsqrtf(var + EPS);
    float ga = gamma[t], fw = fc_w[t];
    avec[t] = ga * inv * fw;
    red[t]  = (beta[t] - ga * inv * mean) * fw;
    __syncthreads();
    for (int s = 256; s > 0; s >>= 1) {
        if (t < s) red[t] += red[t + s];
        __syncthreads();
    }
    if (t == 0) kons[0] = red[0] + fc_b[0];
}

// ---------------------------------------------------------------------------
// Kernel 3: out[b] = sigmoid(K + sum_g a[g] * repsT[g][b])
// grid 16 x 256 threads; coalesced along b
// ---------------------------------------------------------------------------
__global__ void hiscom_out_kernel(const float* __restrict__ repsT,
                                  const float* __restrict__ avec,
                                  const float* __restrict__ kons,
                                  float* __restrict__ out)
{
    const int b = blockIdx.x * blockDim.x + threadIdx.x;
    float acc = kons[0];
#pragma unroll 8
    for (int g = 0; g < GG; ++g) {
        acc += avec[g] * repsT[(size_t)g * BB + b];
    }
    out[b] = 1.0f / (1.0f + expf(-acc));
}

// ---------------------------------------------------------------------------
extern "C" void kernel_launch(void* const* d_in, const int* in_sizes, int n_in,
                              void* d_out, int out_size, void* d_ws, size_t ws_size,
                              hipStream_t stream) {
    (void)in_sizes; (void)n_in; (void)out_size; (void)ws_size;
    const float* x     = (const float*)d_in[0];
    const int*   idx   = (const int*)d_in[1];
    const float* W     = (const float*)d_in[2];
    const float* gamma = (const float*)d_in[3];
    const float* beta  = (const float*)d_in[4];
    const float* fc_w  = (const float*)d_in[5];
    const float* fc_b  = (const float*)d_in[6];
    float* out = (float*)d_out;

    // workspace layout (floats)
    float* wsf    = (float*)d_ws;
    float* repsT  = wsf;                       // G*B = 2,097,152 floats (8 MB)
    float* gsum   = wsf + (size_t)GG * BB;     // 512
    float* gsumsq = gsum + GG;                 // 512
    float* avec   = gsumsq + GG;               // 512
    float* kons   = avec + GG;                 // 1

    // zero the atomic accumulators each call (capture-safe async memset)
    (void)hipMemsetAsync((char*)d_ws + (size_t)GG * BB * sizeof(float), 0,
                         2 * GG * sizeof(float), stream);

    dim3 grid1(GG / 8, BB / 16);
    hipLaunchKernelGGL(hiscom_reps_kernel, grid1, dim3(256), 0, stream,
                       x, idx, W, repsT, gsum, gsumsq);

    hipLaunchKernelGGL(hiscom_stats_kernel, dim3(1), dim3(512), 0, stream,
                       gsum, gsumsq, gamma, beta, fc_w, fc_b, avec, kons);

    hipLaunchKernelGGL(hiscom_out_kernel, dim3(BB / 256), dim3(256), 0, stream,
                       repsT, avec, kons, out);
}